// Model_83889301226199
// MI455X (gfx1250) — compile-verified
//
#include <hip/hip_runtime.h>
#include <math.h>
#include <stdint.h>

typedef __attribute__((ext_vector_type(2))) float v2f;
typedef __attribute__((ext_vector_type(8))) float v8f;

#define SM_SCALE 0.07216878364870322f
#define B_       32
#define HQ       16
#define DN       128
#define DR       64
#define DQ       192    // DN + DR
#define DV       128
#define PAGE     64
#define MAXPAGES 64
#define NSPLIT   16
#define KV_SPLIT 256    // MAXPAGES*PAGE / NSPLIT
#define NW       2      // waves per workgroup
#define KS       196    // padded K row stride, floats (196%64==4 -> conflict-free)
#define VS       136    // padded V row stride, floats (2*136%64==16 -> halves disjoint)
#define KTILE_F  (16 * KS)            // 3136 floats (rope lives at +DN in each row)
#define VTILE_F  (16 * VS)            // 2176 floats
#define STAGE_F  (KTILE_F + VTILE_F)  // 5312 floats per buffer
#define PS       20     // padded P row stride

__device__ __forceinline__ v8f wmma_f32_16x16x4(v2f a, v2f b, v8f c) {
  return __builtin_amdgcn_wmma_f32_16x16x4_f32(
      false, a, false, b, (short)0, c, false, false);
}

// One 512B chunk (32 lanes x b128) global -> LDS, ASYNCcnt-tracked, no VGPR data.
__device__ __forceinline__ void async_b128(uint32_t lds_addr, uint64_t gaddr) {
  asm volatile("global_load_async_to_lds_b128 %0, %1, off"
               :: "v"(lds_addr), "v"(gaddr) : "memory");
}
__device__ __forceinline__ void wait_async_prev40() {  // prev tile's 40 done (in-order)
  asm volatile("s_wait_asynccnt 0x28" ::: "memory");
}
__device__ __forceinline__ void wait_async_all() {
  asm volatile("s_wait_asynccnt 0x0" ::: "memory");
}

// Issue the 40 async b128 copies staging one 16-position KV tile (K-nope 8KB,
// K-rope 4KB, V 8KB -- all contiguous in memory) into padded LDS rows.
__device__ __forceinline__ void issue_tile_async(
    const float* k_nope, const float* k_rope, const float* v_buf,
    size_t rowbase, uint32_t lds_k, uint32_t lds_v,
    uint32_t lb, int col, int hi)
{
  const uint64_t g_kn = (uint64_t)(uintptr_t)(k_nope + rowbase * DN);
  const uint64_t g_kr = (uint64_t)(uintptr_t)(k_rope + rowbase * DR);
  const uint64_t g_v  = (uint64_t)(uintptr_t)(v_buf  + rowbase * DV);
  // K-nope: chunk f=i*32+lane -> row=i, cidx=lane
#pragma unroll
  for (int i = 0; i < 16; ++i)
    async_b128(lds_k + lb + (uint32_t)(i * KS * 4),
               g_kn + lb + (uint64_t)(i * 512));
  // K-rope at +DN within each padded row: row=2i+hi, cidx=col
#pragma unroll
  for (int i = 0; i < 8; ++i)
    async_b128(lds_k + (uint32_t)((2 * i + hi) * KS * 4 + DN * 4 + col * 16),
               g_kr + lb + (uint64_t)(i * 512));
  // V: row=i, cidx=lane
#pragma unroll
  for (int i = 0; i < 16; ++i)
    async_b128(lds_v + lb + (uint32_t)(i * VS * 4),
               g_v + lb + (uint64_t)(i * 512));
}

// ---------------------------------------------------------------------------
// Pass 1: workgroup = (batch, kv-split); 2 waves, each flash-attending
// 16-position KV tiles with double-buffered async global->LDS staging.
// ---------------------------------------------------------------------------
__global__ __launch_bounds__(64) void mla_decode_pass1(
    const float* __restrict__ q,           // [B][HQ][DQ]
    const float* __restrict__ k_nope,      // [NPAGES][PAGE][1][DN]
    const float* __restrict__ k_rope,      // [NPAGES][PAGE][1][DR]
    const float* __restrict__ v_buf,       // [NPAGES][PAGE][1][DV]
    const int*   __restrict__ seq_lens,
    const int*   __restrict__ block_table, // [B][MAXPAGES]
    float* __restrict__ ws_m,              // [B][NSPLIT][HQ]
    float* __restrict__ ws_l,              // [B][NSPLIT][HQ]
    float* __restrict__ ws_o)              // [B][NSPLIT][HQ][DV]
{
  const int b     = blockIdx.x;
  const int split = blockIdx.y;
  const int tid   = threadIdx.x;
  const int wave  = tid >> 5;
  const int lane  = tid & 31;
  const int col   = lane & 15;   // WMMA N index / A row index
  const int hi    = lane >> 4;   // lane half: K-pair 0-1 vs 2-3

  __shared__ float sh_stage[NW][2][STAGE_F]; // per-wave double buffers (o-alias later)
  __shared__ float sh_p[NW][HQ * PS];
  __shared__ float sh_m[NW][HQ];
  __shared__ float sh_l[NW][HQ];

  const uint32_t lds_kb[2] = { (uint32_t)(uintptr_t)&sh_stage[wave][0][0],
                               (uint32_t)(uintptr_t)&sh_stage[wave][1][0] };
  const uint32_t lds_vb[2] = { lds_kb[0] + KTILE_F * 4, lds_kb[1] + KTILE_F * 4 };
  const uint32_t lb = (uint32_t)lane * 16u;

  const int seq      = seq_lens[b];
  const int range_lo = split * KV_SPLIT;
  const int range_hi = min(seq, range_lo + KV_SPLIT);
  const int ntiles   = (range_hi > range_lo) ? ((range_hi - range_lo + 15) >> 4) : 0;
  const int* bt      = block_table + b * MAXPAGES;

  // ---- prologue: start fetching this wave's first tile before anything else ----
  if (wave < ntiles) {
    const int kv0 = range_lo + wave * 16;
    const size_t rowbase = (size_t)bt[kv0 >> 6] * PAGE + (kv0 & (PAGE - 1));
    issue_tile_async(k_nope, k_rope, v_buf, rowbase, lds_kb[0], lds_vb[0], lb, col, hi);
  }

  // ---- Q A-fragments resident in registers (overlaps first async fetch) ----
  const float* qb = q + (size_t)b * HQ * DQ + col * DQ + 2 * hi;
  v2f qa[48];
#pragma unroll
  for (int k = 0; k < 48; ++k) qa[k] = *(const v2f*)(qb + 4 * k);

  float m[8], l[8];
  v8f   o[8];
#pragma unroll
  for (int r = 0; r < 8; ++r) { m[r] = -1e30f; l[r] = 0.0f; }
#pragma unroll
  for (int t = 0; t < 8; ++t)
#pragma unroll
    for (int r = 0; r < 8; ++r) o[t][r] = 0.0f;

  if (wave < ntiles) {
    int pbuf = 0;
    for (int ti = wave; ti < ntiles; ti += NW, pbuf ^= 1) {
      const int kv0 = range_lo + ti * 16;

      // ---- prefetch next tile into the other buffer, then wait current ----
      const int nt = ti + NW;
      if (nt < ntiles) {
        const int nkv0 = range_lo + nt * 16;
        const size_t nrow = (size_t)bt[nkv0 >> 6] * PAGE + (nkv0 & (PAGE - 1));
        issue_tile_async(k_nope, k_rope, v_buf, nrow,
                         lds_kb[pbuf ^ 1], lds_vb[pbuf ^ 1], lb, col, hi);
        wait_async_prev40();   // in-order completion: current buffer is ready
      } else {
        wait_async_all();
      }

      const float* kw = &sh_stage[wave][pbuf][0];
      const float* vw = kw + KTILE_F;

      // ---- scores tile: S(16x16) = Q(16x192) * K^T(192x16) from LDS ----
      v8f c;
#pragma unroll
      for (int r = 0; r < 8; ++r) c[r] = 0.0f;
      const float* krow = kw + col * KS + 2 * hi;
#pragma unroll
      for (int kk = 0; kk < DN / 4; ++kk) {
        v2f bb = *(const v2f*)(krow + 4 * kk);        // ds_load_2addr_b64 pairs
        c = wmma_f32_16x16x4(qa[kk], bb, c);
      }
#pragma unroll
      for (int kk = 0; kk < DR / 4; ++kk) {
        v2f bb = *(const v2f*)(krow + DN + 4 * kk);
        c = wmma_f32_16x16x4(qa[32 + kk], bb, c);
      }

      // ---- online softmax (row = head, across 16 lanes of the half) ----
      const bool valid = (kv0 + col) < seq;
      float p[8];
#pragma unroll
      for (int r = 0; r < 8; ++r) {
        float s = valid ? c[r] * SM_SCALE : -1e30f;
        float tm = s;
#pragma unroll
        for (int off = 8; off >= 1; off >>= 1)
          tm = fmaxf(tm, __shfl_xor(tm, off, 16));
        float mn = fmaxf(m[r], tm);
        float al = __expf(m[r] - mn);
        float pv = __expf(s - mn);
        float rs = pv;
#pragma unroll
        for (int off = 8; off >= 1; off >>= 1)
          rs += __shfl_xor(rs, off, 16);
        l[r] = l[r] * al + rs;
        m[r] = mn;
        p[r] = pv;
#pragma unroll
        for (int t = 0; t < 8; ++t) o[t][r] *= al;
      }

      // ---- transpose P (C layout -> A layout) through per-wave LDS ----
      float* pw = &sh_p[wave][0];
#pragma unroll
      for (int r = 0; r < 8; ++r)
        pw[(8 * hi + r) * PS + col] = p[r];

      // ---- O(16x128) += P(16x16) * V(16x128), V from LDS ----
      const float* prow = pw + col * PS + 2 * hi;
#pragma unroll
      for (int j = 0; j < 4; ++j) {
        v2f a = *(const v2f*)(prow + 4 * j);
        const float* vp = vw + (4 * j + 2 * hi) * VS + col;
#pragma unroll
        for (int t = 0; t < 8; ++t) {
          v2f bb;
          bb.x = vp[16 * t];
          bb.y = vp[16 * t + VS];
          o[t] = wmma_f32_16x16x4(a, bb, o[t]);
        }
      }
    }
  }

  // ---- publish per-wave state (accumulators alias staging buffer 0) ----
  if (col == 0) {
#pragma unroll
    for (int r = 0; r < 8; ++r) {
      sh_m[wave][8 * hi + r] = m[r];
      sh_l[wave][8 * hi + r] = l[r];
    }
  }
  float* ow = &sh_stage[wave][0][0];
#pragma unroll
  for (int t = 0; t < 8; ++t)
#pragma unroll
    for (int r = 0; r < 8; ++r)
      ow[(8 * hi + r) * DV + 16 * t + col] = o[t][r];
  __syncthreads();

  // ---- combine NW waves, write split partial ----
  float* wso = ws_o + ((size_t)b * NSPLIT + split) * HQ * DV;
#pragma unroll
  for (int i = 0; i < 32; ++i) {
    int e = tid + i * 64;               // 0 .. 2047
    int row = e >> 7, cc = e & 127;
    float M = fmaxf(sh_m[0][row], sh_m[1][row]);
    float acc = 0.0f;
#pragma unroll
    for (int w = 0; w < NW; ++w)
      acc += __expf(sh_m[w][row] - M) * sh_stage[w][0][row * DV + cc];
    wso[row * DV + cc] = acc;
  }
  if (tid < HQ) {
    int row = tid;
    float M = fmaxf(sh_m[0][row], sh_m[1][row]);
    float L = 0.0f;
#pragma unroll
    for (int w = 0; w < NW; ++w)
      L += __expf(sh_m[w][row] - M) * sh_l[w][row];
    ws_m[((size_t)b * NSPLIT + split) * HQ + row] = M;
    ws_l[((size_t)b * NSPLIT + split) * HQ + row] = L;
  }
}

// ---------------------------------------------------------------------------
// Pass 2: combine NSPLIT partials per (batch, head) and normalize.
// ---------------------------------------------------------------------------
__global__ __launch_bounds__(128) void mla_decode_pass2(
    const float* __restrict__ ws_m,
    const float* __restrict__ ws_l,
    const float* __restrict__ ws_o,
    float* __restrict__ out)               // [B][HQ][DV]
{
  const int b = blockIdx.x;
  const int h = blockIdx.y;
  const int d = threadIdx.x;               // 0..127

  float ms[NSPLIT];
  float M = -1e30f;
#pragma unroll
  for (int s = 0; s < NSPLIT; ++s) {
    ms[s] = ws_m[((size_t)b * NSPLIT + s) * HQ + h];
    M = fmaxf(M, ms[s]);
  }
  float L = 0.0f, acc = 0.0f;
#pragma unroll
  for (int s = 0; s < NSPLIT; ++s) {
    float f = __expf(ms[s] - M);
    L   += f * ws_l[((size_t)b * NSPLIT + s) * HQ + h];
    acc += f * ws_o[(((size_t)b * NSPLIT + s) * HQ + h) * DV + d];
  }
  out[((size_t)b * HQ + h) * DV + d] = (L > 0.0f) ? (acc / L) : 0.0f;
}

// ---------------------------------------------------------------------------
extern "C" void kernel_launch(void* const* d_in, const int* in_sizes, int n_in,
                              void* d_out, int out_size, void* d_ws, size_t ws_size,
                              hipStream_t stream) {
  const float* q       = (const float*)d_in[0];
  const float* k_nope  = (const float*)d_in[1];
  const float* k_rope  = (const float*)d_in[2];
  const float* v_buf   = (const float*)d_in[3];
  const int*   seqs    = (const int*)d_in[4];
  const int*   btable  = (const int*)d_in[5];
  float*       out     = (float*)d_out;

  float* ws_m = (float*)d_ws;                       // B*NSPLIT*HQ
  float* ws_l = ws_m + (size_t)B_ * NSPLIT * HQ;
  float* ws_o = ws_l + (size_t)B_ * NSPLIT * HQ;    // B*NSPLIT*HQ*DV

  mla_decode_pass1<<<dim3(B_, NSPLIT), 64, 0, stream>>>(
      q, k_nope, k_rope, v_buf, seqs, btable, ws_m, ws_l, ws_o);
  mla_decode_pass2<<<dim3(B_, HQ), 128, 0, stream>>>(
      ws_m, ws_l, ws_o, out);
}